// RegressionParser_91122026151984
// MI455X (gfx1250) — compile-verified
//
#include <hip/hip_runtime.h>
#include <math.h>

// RetinaFace decode + greedy NMS (750 picks) + gated gather.
// N_PR = 2*(320^2 + 160^2 + 80^2) = 268800 priors.
#define N_PR    268800
#define L0_END  204800      // step 8,  f=320, min_sizes {16,32}
#define L1_END  256000      // step 16, f=160, min_sizes {64,128}
#define NMS_KEEP 750
#define RF_IOU  0.4f

// gfx1250 async-copy path (upstream clang builtins); fall back to plain loads.
#if __has_builtin(__builtin_amdgcn_global_load_async_to_lds_b128) && \
    __has_builtin(__builtin_amdgcn_s_wait_asynccnt)
#define RF_HAVE_ASYNC_LDS 1
// Builtin signature (from hipcc diagnostic): void(v4i AS1*, v4i AS3*, Imm, Imm)
typedef int rf_v4i __attribute__((vector_size(16)));
typedef __attribute__((address_space(1))) rf_v4i* rf_gv4i_p;
typedef __attribute__((address_space(3))) rf_v4i* rf_lv4i_p;
#endif

// ---------------------------------------------------------------- priors ----
// numpy builds priors in float64 and casts to float32 at the end; reproduce.
__device__ __forceinline__ void rf_prior(int j, float& pcx, float& pcy, float& ps) {
  int r = j, f, step;
  double msz;
  if (r < L0_END)       { f = 320; step = 8;  msz = (r & 1) ? 32.0  : 16.0;  }
  else if (r < L1_END)  { r -= L0_END; f = 160; step = 16; msz = (r & 1) ? 128.0 : 64.0;  }
  else                  { r -= L1_END; f = 80;  step = 32; msz = (r & 1) ? 512.0 : 256.0; }
  int cell = r >> 1;
  int x = cell % f, y = cell / f;
  pcx = (float)((((double)x + 0.5) * (double)step) / 2560.0);
  pcy = (float)((((double)y + 0.5) * (double)step) / 2560.0);
  ps  = (float)(msz / 2560.0);
}

// Monotone 64-bit key: higher score wins; ties -> smaller original index wins
// (matches stable argsort of -scores followed by first-True argmax).
__device__ __forceinline__ unsigned long long rf_pack(float s, int j) {
  unsigned u = __float_as_uint(s);
  u = (u & 0x80000000u) ? ~u : (u | 0x80000000u);
  return ((unsigned long long)u << 32) | (unsigned long long)(0x7FFFFFFFu - (unsigned)j);
}
__device__ __forceinline__ unsigned long long rf_max64(unsigned long long a, unsigned long long b) {
  return a > b ? a : b;
}

// ------------------------------------------------------------------ init ----
__global__ void rf_init_keys(unsigned long long* __restrict__ keys) {
  int i = blockIdx.x * 256 + threadIdx.x;
  if (i <= NMS_KEEP) keys[i] = 0ULL;   // keys[0..750]
}

// ---------------------------------------------------------------- decode ----
// boxes[j] = decoded (x1,y1,x2,y2)*2560; wscore = working copy of scores;
// keys[0] = argmax key over all scores (seed pivot for NMS round 0).
__global__ void rf_decode(const float4* __restrict__ loc,
                          const float*  __restrict__ scores,
                          float4* __restrict__ boxes,
                          float*  __restrict__ wscore,
                          unsigned long long* __restrict__ keys) {
  __shared__ unsigned long long red[256];
  int j = blockIdx.x * 256 + threadIdx.x;
  unsigned long long local = 0ULL;
  if (j < N_PR) {
    float pcx, pcy, ps;
    rf_prior(j, pcx, pcy, ps);
    float4 l = loc[j];
    // JAX f32 op order: priors[:,:2] + (loc[:,:2]*0.1)*priors[:,2:]
    float cx = pcx + (l.x * 0.1f) * ps;
    float cy = pcy + (l.y * 0.1f) * ps;
    float w  = ps * expf(l.z * 0.2f);
    float h  = ps * expf(l.w * 0.2f);
    float4 b;
    b.x = (cx - w * 0.5f) * 2560.0f;
    b.y = (cy - h * 0.5f) * 2560.0f;
    b.z = (cx + w * 0.5f) * 2560.0f;
    b.w = (cy + h * 0.5f) * 2560.0f;
    boxes[j]  = b;
    float s   = scores[j];
    wscore[j] = s;
    local = rf_pack(s, j);
  }
  red[threadIdx.x] = local;
  __syncthreads();
  for (int st = 128; st > 0; st >>= 1) {
    if (threadIdx.x < st) red[threadIdx.x] = rf_max64(red[threadIdx.x], red[threadIdx.x + st]);
    __syncthreads();
  }
  if (threadIdx.x == 0 && red[0]) atomicMax(&keys[0], red[0]);
}

// -------------------------------------------------------------- NMS round ----
// One surviving element: IOU vs pivot, suppress or contribute to next argmax.
__device__ __forceinline__ void rf_proc(float& s, int j,
                                        const float4* __restrict__ boxes,
                                        float4 pb, float parea,
                                        unsigned long long& local, bool& changed) {
  if (s != -INFINITY) {
    float4 b    = boxes[j];
    float xx1   = fmaxf(pb.x, b.x);
    float yy1   = fmaxf(pb.y, b.y);
    float xx2   = fminf(pb.z, b.z);
    float yy2   = fminf(pb.w, b.w);
    float inter = fmaxf(0.0f, xx2 - xx1 + 1.0f) * fmaxf(0.0f, yy2 - yy1 + 1.0f);
    float area  = (b.z - b.x + 1.0f) * (b.w - b.y + 1.0f);
    float iou   = inter / (parea + area - inter);
    if (iou > RF_IOU) { s = -INFINITY; changed = true; }  // pivot self-suppresses (iou==1)
    else              local = rf_max64(local, rf_pack(s, j));
  }
}

// Fused: suppress vs. pivot keys[iter], and max-reduce surviving scores into
// keys[iter+1]. Each thread owns 4 consecutive priors (one b128 score quad);
// box loads only for survivors, score quad written back only if changed.
// Stream ordering of the 750 launches provides the global barrier.
__global__ void rf_nms_iter(const float4* __restrict__ boxes,
                            float4* __restrict__ wscore4,
                            unsigned long long* __restrict__ keys,
                            int*    __restrict__ keep,
                            int iter) {
  __shared__ unsigned long long red[256];
  __shared__ float4 s_pivot;
  unsigned long long pk = keys[iter];
  int tid = blockIdx.x * 256 + threadIdx.x;
  if (pk == 0ULL) {                       // nothing survives: cascade -1s
    if (tid == 0) keep[iter] = -1;
    return;
  }
  int pidx = (int)(0x7FFFFFFFu - (unsigned)(pk & 0xFFFFFFFFull));
  if (tid == 0) keep[iter] = pidx;

#ifdef RF_HAVE_ASYNC_LDS
  // gfx1250: lane 0 async-copies the pivot box global->LDS, waits ASYNCcnt,
  // then every wave reads it from LDS (uniform broadcast without per-wave
  // global loads): global_load_async_to_lds_b128 + s_wait_asynccnt + ds_load.
  if (threadIdx.x == 0) {
    __builtin_amdgcn_global_load_async_to_lds_b128(
        (rf_gv4i_p)(boxes + pidx), (rf_lv4i_p)&s_pivot, 0, 0);
    __builtin_amdgcn_s_wait_asynccnt(0);
  }
  __syncthreads();
  float4 pb = s_pivot;
#else
  float4 pb = boxes[pidx];
  (void)s_pivot;
#endif
  float parea = (pb.z - pb.x + 1.0f) * (pb.w - pb.y + 1.0f);

  unsigned long long local = 0ULL;
  if (tid < N_PR / 4) {
    int j = tid * 4;
    // gfx1250: global_prefetch_b8 the box lines for this quad while the
    // score quad load is in flight.
    __builtin_prefetch(&boxes[j], 0, 0);
    __builtin_prefetch(&boxes[j + 3], 0, 0);
    float4 s4 = wscore4[tid];
    bool changed = false;
    rf_proc(s4.x, j + 0, boxes, pb, parea, local, changed);
    rf_proc(s4.y, j + 1, boxes, pb, parea, local, changed);
    rf_proc(s4.z, j + 2, boxes, pb, parea, local, changed);
    rf_proc(s4.w, j + 3, boxes, pb, parea, local, changed);
    if (changed) wscore4[tid] = s4;
  }
  red[threadIdx.x] = local;
  __syncthreads();
  for (int st = 128; st > 0; st >>= 1) {
    if (threadIdx.x < st) red[threadIdx.x] = rf_max64(red[threadIdx.x], red[threadIdx.x + st]);
    __syncthreads();
  }
  if (threadIdx.x == 0 && red[0]) atomicMax(&keys[iter + 1], red[0]);
}

// ---------------------------------------------------------------- gather ----
// Landmarks are decoded lazily here (only 750 needed) instead of storing 10.8MB.
__global__ void rf_gather(const int* __restrict__ keep,
                          const float4* __restrict__ boxes,
                          const float* __restrict__ scores,
                          const float* __restrict__ landms,
                          const float* __restrict__ thr,
                          float* __restrict__ out) {
  int i = blockIdx.x * 256 + threadIdx.x;
  if (i >= NMS_KEEP) return;
  int idx   = keep[i];
  float t   = thr[0];
  float sc  = 0.0f;
  bool valid = idx >= 0;
  if (valid) { sc = scores[idx]; valid = sc > t; }

  float* bo = out;                   // [750,4]
  float* so = out + NMS_KEEP * 4;    // [750]
  float* lo = out + NMS_KEEP * 5;    // [750,10]

  if (valid) {
    float4 b = boxes[idx];
    bo[i * 4 + 0] = b.x; bo[i * 4 + 1] = b.y;
    bo[i * 4 + 2] = b.z; bo[i * 4 + 3] = b.w;
    so[i] = sc;
    float pcx, pcy, ps;
    rf_prior(idx, pcx, pcy, ps);
#pragma unroll
    for (int k = 0; k < 5; ++k) {
      float lx = landms[idx * 10 + 2 * k];
      float ly = landms[idx * 10 + 2 * k + 1];
      lo[i * 10 + 2 * k]     = (pcx + (lx * 0.1f) * ps) * 2560.0f;
      lo[i * 10 + 2 * k + 1] = (pcy + (ly * 0.1f) * ps) * 2560.0f;
    }
  } else {
    bo[i * 4 + 0] = 0.0f; bo[i * 4 + 1] = 0.0f;
    bo[i * 4 + 2] = 0.0f; bo[i * 4 + 3] = 0.0f;
    so[i] = 0.0f;
#pragma unroll
    for (int k = 0; k < 10; ++k) lo[i * 10 + k] = 0.0f;
  }
}

// ---------------------------------------------------------------- launch ----
extern "C" void kernel_launch(void* const* d_in, const int* in_sizes, int n_in,
                              void* d_out, int out_size, void* d_ws, size_t ws_size,
                              hipStream_t stream) {
  const float* bboxes = (const float*)d_in[0];  // [1, N, 4]
  const float* scores = (const float*)d_in[1];  // [1, N]
  const float* landms = (const float*)d_in[2];  // [1, N, 10]
  const float* thr    = (const float*)d_in[3];  // [1]

  // Workspace layout (~5.14 MB; everything L2-resident):
  char* ws = (char*)d_ws;
  float4* boxes = (float4*)ws;                                   // N*16 B
  float*  wscr  = (float*)(ws + (size_t)N_PR * 16);              // N*4 B (16B-aligned)
  unsigned long long* keys =
      (unsigned long long*)(ws + (size_t)N_PR * 20);             // 751*8 B
  int* keep = (int*)(ws + (size_t)N_PR * 20 + (NMS_KEEP + 1) * 8); // 750*4 B

  rf_init_keys<<<3, 256, 0, stream>>>(keys);
  rf_decode<<<N_PR / 256, 256, 0, stream>>>((const float4*)bboxes, scores,
                                            boxes, wscr, keys);
  const int nms_blocks = (N_PR / 4 + 255) / 256;   // 263
  for (int i = 0; i < NMS_KEEP; ++i)
    rf_nms_iter<<<nms_blocks, 256, 0, stream>>>(boxes, (float4*)wscr, keys, keep, i);
  rf_gather<<<3, 256, 0, stream>>>(keep, boxes, scores, landms, thr, (float*)d_out);
}